// PN_Plus_VAE_15333033247098
// MI455X (gfx1250) — compile-verified
//
#include <hip/hip_runtime.h>
#include <hip/hip_bf16.h>

// ---------------------------------------------------------------------------
// PN+VAE fused pipeline for MI455X (gfx1250), wave32, f32 WMMA 16x16x4.
//
// Algebraic reduction:  pooled = relu( (mask*x) @ G + W1[11,:]*(mask@b_param)
//                                      + b1 * rowsum(mask) )
// with G[d,k] = W1[0,k] + sum_j F[d,j]*W1[j+1,k].
// Then MLP 20->500->200->20 (all WMMA, zero-padded to multiples of 16).
// Memory-bound on x+mask (32 MB @ 23.3 TB/s ~ 1.4us); weights stay in L2.
// ---------------------------------------------------------------------------

typedef __attribute__((ext_vector_type(2))) float v2f;
typedef __attribute__((ext_vector_type(8))) float v8f;

#define BSZ   4096
#define DDIM  1024

// f32 workspace layout (element offsets)
constexpr int GP_OFF  = 0;        // G  padded 1024 x 32
constexpr int W2P_OFF = 32768;    // W2 padded   32 x 512
constexpr int W3P_OFF = 49152;    // W3 padded  512 x 208
constexpr int W4P_OFF = 155648;   // W4 padded  208 x 32
constexpr int B2P_OFF = 162304;   // 512
constexpr int B3P_OFF = 162816;   // 208
constexpr int B4P_OFF = 163024;   // 32
// total 163056 floats = 652,224 bytes of d_ws

__device__ __forceinline__ v8f wmma4(v2f a, v2f b, v8f c) {
  // D = A(16x4) * B(4x16) + C, f32, wave32
  return __builtin_amdgcn_wmma_f32_16x16x4_f32(
      /*neg_a=*/false, a, /*neg_b=*/false, b,
      /*c_mod=*/(short)0, c, /*reuse_a=*/false, /*reuse_b=*/false);
}

// ---------------------------------------------------------------------------
// Prep: fold F/W1 into G; zero-pad weights/biases into WMMA-friendly shapes.
// ---------------------------------------------------------------------------
__global__ void prep_kernel(const float* __restrict__ F,
                            const float* __restrict__ W1,
                            const float* __restrict__ W2,
                            const float* __restrict__ W3,
                            const float* __restrict__ W4,
                            const float* __restrict__ b2,
                            const float* __restrict__ b3,
                            const float* __restrict__ b4,
                            float* __restrict__ ws) {
  const int i = blockIdx.x * blockDim.x + threadIdx.x;

  if (i < 1024 * 32) {                       // G padded (1024 x 32)
    const int d = i >> 5, k = i & 31;
    float v = 0.f;
    if (k < 20) {
      v = W1[k];                             // W1[0, k]
      #pragma unroll
      for (int j = 0; j < 10; ++j) v += F[d * 10 + j] * W1[(j + 1) * 20 + k];
    }
    ws[GP_OFF + i] = v;
  }
  if (i < 32 * 512) {                        // W2 padded (32 x 512)
    const int k = i >> 9, n = i & 511;
    ws[W2P_OFF + i] = (k < 20 && n < 500) ? W2[k * 500 + n] : 0.f;
  }
  if (i < 512 * 208) {                       // W3 padded (512 x 208)
    const int k = i / 208, n = i % 208;
    ws[W3P_OFF + i] = (k < 500 && n < 200) ? W3[k * 200 + n] : 0.f;
  }
  if (i < 208 * 32) {                        // W4 padded (208 x 32)
    const int k = i >> 5, n = i & 31;
    ws[W4P_OFF + i] = (k < 200 && n < 20) ? W4[k * 20 + n] : 0.f;
  }
  if (i < 512) ws[B2P_OFF + i] = (i < 500) ? b2[i] : 0.f;
  if (i < 208) ws[B3P_OFF + i] = (i < 200) ? b3[i] : 0.f;
  if (i < 32)  ws[B4P_OFF + i] = (i < 20)  ? b4[i] : 0.f;
}

// ---------------------------------------------------------------------------
// Fused main kernel: one workgroup (4 wave32s) carries a 16-row batch stripe
// through GEMM(pooled) -> MLP -> output, intermediates in LDS.
// ---------------------------------------------------------------------------
__global__ __launch_bounds__(128) void vae_main(
    const float* __restrict__ x, const float* __restrict__ mask,
    const float* __restrict__ bparam, const float* __restrict__ W1,
    const float* __restrict__ b1, const float* __restrict__ ws,
    float* __restrict__ out) {
  // padded inner dims (+4 floats) -> conflict-free column reads of A frags
  __shared__ float redC[4][16][36];   // per-wave partial C tiles (stage A)
  __shared__ float sp[16][36];        // pooled (16 x 32)
  __shared__ float h1[16][516];       // hidden1 (16 x 512)
  __shared__ float h2[16][212];       // hidden2 (16 x 208)
  __shared__ float S1s[16], S2s[16];  // rowsum(mask), mask@b_param

  const int tid   = threadIdx.x;
  const int w     = tid >> 5;     // wave id 0..3 (wave32)
  const int lane  = tid & 31;
  const int lrow  = lane & 15;    // A-frag row / B-frag column
  const int lhalf = lane >> 4;    // K sub-pair selector
  const int rb    = blockIdx.x * 16;

  if (tid < 16) { S1s[tid] = 0.f; S2s[tid] = 0.f; }
  __syncthreads();

  const float* __restrict__ Gp = ws + GP_OFF;

  // ---- Stage A: pooled_pre = (mask*x) @ G, K=1024 split across 4 waves ----
  v8f acc0 = {}; v8f acc1 = {};
  float s1 = 0.f, s2 = 0.f;
  const int  kbase  = w * 256;
  const long rowoff = (long)(rb + lrow) * DDIM;
  for (int s = 0; s < 64; ++s) {
    const int kk = kbase + s * 4 + 2 * lhalf;       // K pair base for this lane
    const float2 xv = *(const float2*)(x + rowoff + kk);
    const float2 mv = *(const float2*)(mask + rowoff + kk);
    const float2 bv = *(const float2*)(bparam + kk);
    v2f a; a.x = xv.x * mv.x; a.y = xv.y * mv.y;
    s1 += mv.x + mv.y;
    s2 += mv.x * bv.x + mv.y * bv.y;
    v2f b0, b1f;
    b0.x  = Gp[kk * 32 + lrow];        b0.y  = Gp[(kk + 1) * 32 + lrow];
    b1f.x = Gp[kk * 32 + 16 + lrow];   b1f.y = Gp[(kk + 1) * 32 + 16 + lrow];
    acc0 = wmma4(a, b0, acc0);         // N tile 0..15
    acc1 = wmma4(a, b1f, acc1);        // N tile 16..31
  }
  #pragma unroll
  for (int v = 0; v < 8; ++v) {
    redC[w][v + 8 * lhalf][lrow]      = acc0[v];
    redC[w][v + 8 * lhalf][16 + lrow] = acc1[v];
  }
  atomicAdd(&S1s[lrow], s1);   // lanes r and r+16 of 4 waves cover all 1024 cols
  atomicAdd(&S2s[lrow], s2);
  __syncthreads();

  // cross-wave reduce + correction terms + ReLU -> pooled (zero-padded cols)
  for (int e = tid; e < 512; e += 128) {
    const int r = e >> 5, n = e & 31;
    float pooled = 0.f;
    if (n < 20) {
      float v = redC[0][r][n] + redC[1][r][n] + redC[2][r][n] + redC[3][r][n];
      v += W1[11 * 20 + n] * S2s[r] + b1[n] * S1s[r];
      pooled = v > 0.f ? v : 0.f;
    }
    sp[r][n] = pooled;
  }
  __syncthreads();

  // ---- Stage B: h1 = relu(pooled @ W2p + b2), 16x512, K=32 ----------------
  const float* __restrict__ W2p = ws + W2P_OFF;
  const float* __restrict__ b2p = ws + B2P_OFF;
  for (int t = 0; t < 8; ++t) {
    const int n0 = (w * 8 + t) * 16;
    v8f acc = {};
    #pragma unroll
    for (int s = 0; s < 8; ++s) {
      const int kk = s * 4 + 2 * lhalf;
      v2f a; a.x = sp[lrow][kk];                  a.y = sp[lrow][kk + 1];
      v2f b; b.x = W2p[kk * 512 + n0 + lrow];     b.y = W2p[(kk + 1) * 512 + n0 + lrow];
      acc = wmma4(a, b, acc);
    }
    const float bias = b2p[n0 + lrow];
    #pragma unroll
    for (int v = 0; v < 8; ++v) {
      const float val = acc[v] + bias;
      h1[v + 8 * lhalf][n0 + lrow] = val > 0.f ? val : 0.f;
    }
  }
  __syncthreads();

  // ---- Stage C: h2 = relu(h1 @ W3p + b3), 16x208, K=512 -------------------
  const float* __restrict__ W3p = ws + W3P_OFF;
  const float* __restrict__ b3p = ws + B3P_OFF;
  for (int nt = w; nt < 13; nt += 4) {
    const int n0 = nt * 16;
    v8f acc = {};
    #pragma unroll 4
    for (int s = 0; s < 128; ++s) {
      const int kk = s * 4 + 2 * lhalf;
      v2f a; a.x = h1[lrow][kk];                  a.y = h1[lrow][kk + 1];
      v2f b; b.x = W3p[kk * 208 + n0 + lrow];     b.y = W3p[(kk + 1) * 208 + n0 + lrow];
      acc = wmma4(a, b, acc);
    }
    const float bias = b3p[n0 + lrow];
    #pragma unroll
    for (int v = 0; v < 8; ++v) {
      const float val = acc[v] + bias;
      h2[v + 8 * lhalf][n0 + lrow] = val > 0.f ? val : 0.f;
    }
  }
  __syncthreads();

  // ---- Stage D: out = h2 @ W4p + b4, 16x32 (only cols 0..19 real) ---------
  const float* __restrict__ W4p = ws + W4P_OFF;
  const float* __restrict__ b4p = ws + B4P_OFF;
  if (w < 2) {                        // wave-uniform: EXEC all-1 inside
    const int n0 = w * 16;
    v8f acc = {};
    #pragma unroll 4
    for (int s = 0; s < 52; ++s) {
      const int kk = s * 4 + 2 * lhalf;
      v2f a; a.x = h2[lrow][kk];                  a.y = h2[lrow][kk + 1];
      v2f b; b.x = W4p[kk * 32 + n0 + lrow];      b.y = W4p[(kk + 1) * 32 + n0 + lrow];
      acc = wmma4(a, b, acc);
    }
    const int n = n0 + lrow;
    const float bias = b4p[n];
    #pragma unroll
    for (int v = 0; v < 8; ++v) {
      const int brow = rb + v + 8 * lhalf;
      const float val = acc[v] + bias;
      if (n < 10)       out[brow * 10 + n] = val;                    // mean
      else if (n < 20)  out[40960 + brow * 10 + (n - 10)] = val;     // logvar
    }
  }
}

// ---------------------------------------------------------------------------
extern "C" void kernel_launch(void* const* d_in, const int* in_sizes, int n_in,
                              void* d_out, int out_size, void* d_ws, size_t ws_size,
                              hipStream_t stream) {
  const float* x      = (const float*)d_in[0];
  const float* mask   = (const float*)d_in[1];
  const float* F      = (const float*)d_in[2];
  const float* bparam = (const float*)d_in[3];
  const float* W1     = (const float*)d_in[4];
  const float* b1     = (const float*)d_in[5];
  const float* W2     = (const float*)d_in[6];
  const float* b2     = (const float*)d_in[7];
  const float* W3     = (const float*)d_in[8];
  const float* b3     = (const float*)d_in[9];
  const float* W4     = (const float*)d_in[10];
  const float* b4     = (const float*)d_in[11];
  float* ws  = (float*)d_ws;
  float* out = (float*)d_out;

  prep_kernel<<<(512 * 208 + 255) / 256, 256, 0, stream>>>(
      F, W1, W2, W3, W4, b2, b3, b4, ws);
  vae_main<<<BSZ / 16, 128, 0, stream>>>(x, mask, bparam, W1, b1, ws, out);
}